// LearnableCanny_80564996538465
// MI455X (gfx1250) — compile-verified
//
#include <hip/hip_runtime.h>
#include <math.h>

#define HH 512
#define WW 512
#define BB 8
#define CHW (HH*WW)          /* 262144 */
#define NTOT (BB*CHW)        /* 2097152 */
#define PI_F 3.14159265358979323846f
#define NBINS 2048
#define KTARGET ((CHW-1)/2)  /* 131071 */

/* aux layout (32-bit words at start of workspace) */
#define AX_S      0
#define AX_W      1
#define AX_MED    2
#define AX_LOW    3
#define AX_HIGH   4
#define AX_MXKEY  5
#define AX_MINKEY 6
#define AX_MAXKEY 7
#define AX_LO     8    /* float[8]  */
#define AX_WD     16   /* float[8]  */
#define AX_K      24   /* uint[8]   */
#define AX_HIST   64   /* uint[8*2048] */
#define BIGOFF    (1<<17) /* big buffers start at 128KB */

typedef __attribute__((ext_vector_type(2))) float v2f;
typedef __attribute__((ext_vector_type(8))) float v8f;

/* taps t = u*3+v -> offset (u-1, v-1); t=9..11 are K-padding (coef 0).
 * constexpr so all indexing with unroll-constant t folds to immediates. */
__device__ constexpr float FC[4][12] = {
  {-1, 0, 1, -2, 0, 2, -1, 0, 1, 0, 0, 0},   /* KX   */
  {-1,-2,-1,  0, 0, 0,  1, 2, 1, 0, 0, 0},   /* KY   */
  {-2,-1, 0, -1, 0, 1,  0, 1, 2, 0, 0, 0},   /* K45  */
  { 0, 1, 2, -1, 0, 1, -2,-1, 0, 0, 0, 0},   /* K135 */
};
__device__ constexpr int TDI[12] = {-1,-1,-1, 0,0,0, 1,1,1, 0,0,0};
__device__ constexpr int TDJ[12] = {-1, 0, 1,-1,0,1,-1,0,1, 0,0,0};

__device__ __forceinline__ int clampi(int v, int lo, int hi){ return v<lo?lo:(v>hi?hi:v); }
__device__ __forceinline__ unsigned okey(float f){
  unsigned u = __float_as_uint(f);
  return (u & 0x80000000u) ? ~u : (u | 0x80000000u);
}
__device__ __forceinline__ float keyf(unsigned k){
  unsigned u = (k & 0x80000000u) ? (k & 0x7FFFFFFFu) : ~k;
  return __uint_as_float(u);
}
__device__ __forceinline__ float hsw(float x){
  float t = fminf(fmaxf(x + 3.0f, 0.0f), 6.0f);
  return x * t * (1.0f/6.0f);
}
/* branchless zero-padded read: always loads from a clamped (legal) address,
 * selects 0 when (ii,jj) is outside the image -> no exec-mask branching. */
__device__ __forceinline__ float ldz(const float* __restrict__ ch, int ii, int jj){
  float v = ch[clampi(ii,0,511)*512 + clampi(jj,0,511)];
  bool inr = ((unsigned)ii < 512u) && ((unsigned)jj < 512u);
  return inr ? v : 0.0f;
}
/* immediate-select of filter coefficient: column n (0..15), tap t compile-time */
__device__ __forceinline__ float bcoef(int n, float c0, float c1, float c2, float c3){
  return (n==0)? c0 : (n==1)? c1 : (n==2)? c2 : (n==3)? c3 : 0.0f;
}

/* ---------------- per-image scalar prep ---------------- */
__global__ void k_prep(const float* __restrict__ sigma, const float* __restrict__ weight,
                       float* auxf){
  unsigned* auxu = (unsigned*)auxf;
  int t = threadIdx.x;
  if (t == 0){
    auxf[AX_S] = 1.0f - 1.0f/(1.0f + expf(-sigma[0]));
    auxf[AX_W] = 1.0f/(1.0f + expf(-weight[0]));
    auxu[AX_MXKEY]  = okey(0.0f);
    auxu[AX_MINKEY] = 0xFFFFFFFFu;
    auxu[AX_MAXKEY] = 0u;
  }
  if (t < 8){
    auxf[AX_LO + t] = 0.0f;
    auxf[AX_WD + t] = 8.0f;
    auxu[AX_K  + t] = (unsigned)KTARGET;
  }
}

/* ---------------- adaptive smoothing ---------------- */
__global__ void k_smooth_w(const float* __restrict__ f, float* __restrict__ w){
  int p = blockIdx.x*256 + threadIdx.x;
  int b = p >> 18, i = (p >> 9) & 511, j = p & 511;
  const float* ch = f + ((size_t)b << 18);
  float up = ch[clampi(i-1,0,511)*512 + j];
  float dn = ch[clampi(i+1,0,511)*512 + j];
  float lf = ch[i*512 + clampi(j-1,0,511)];
  float rt = ch[i*512 + clampi(j+1,0,511)];
  float gx = (dn - up)*0.5f, gy = (rt - lf)*0.5f;
  w[p] = expf(-(gx*gx + gy*gy) * (1.0f/200.0f));   /* 2*k*k, k=10 */
}

__global__ void k_smooth_f(const float* __restrict__ f, const float* __restrict__ w,
                           float* __restrict__ o){
  int p = blockIdx.x*256 + threadIdx.x;
  int b = p >> 18, i = (p >> 9) & 511, j = p & 511;
  const float* fc = f + ((size_t)b << 18);
  const float* wc = w + ((size_t)b << 18);
  float num = 0.0f, den = 0.0f;
  #pragma unroll
  for (int a = -1; a <= 1; ++a){
    int ii = clampi(i+a,0,511);
    #pragma unroll
    for (int c = -1; c <= 1; ++c){
      int jj = clampi(j+c,0,511);
      float wv = wc[ii*512 + jj];
      num += fc[ii*512 + jj] * wv;
      den += wv;
    }
  }
  o[p] = num / fmaxf(den, 1e-8f);
}

/* -------- directional 3x3 conv bank via V_WMMA_F32_16X16X4_F32 --------
 * D(16px x 16filt) = A(16px x 12taps) * B(12taps x 16filt), chained k=4.
 * B is compile-time: materialized as immediate cndmask chains (no memory).
 * Outputs: mag, ang (atan2 mod pi), sob = 0.5(|ex|+|ey|).
 */
__global__ void k_dir_wmma(const float* __restrict__ blurred,
                           float* __restrict__ mag, float* __restrict__ ang,
                           float* __restrict__ sob){
  __shared__ float sm[8][32][16];
  const int tid   = threadIdx.x;
  const int lane  = tid & 31, wave = tid >> 5;
  const int wavebase = blockIdx.x*256 + wave*32;
  const int rowM  = lane & 15;
  const bool lowh = (lane < 16);      /* low K-half: taps 4c..4c+1; high: 4c+2..4c+3 */
  const int ncol  = lane & 15;

  /* B vectors: loop-invariant, pure immediates selected by (ncol, lowh) */
  v2f bv[3];
  #pragma unroll
  for (int c = 0; c < 3; ++c){
    const int tl0 = 4*c,     tl1 = 4*c + 1;   /* low-half taps  */
    const int th0 = 4*c + 2, th1 = 4*c + 3;   /* high-half taps */
    float bl0 = bcoef(ncol, FC[0][tl0], FC[1][tl0], FC[2][tl0], FC[3][tl0]);
    float bl1 = bcoef(ncol, FC[0][tl1], FC[1][tl1], FC[2][tl1], FC[3][tl1]);
    float bh0 = bcoef(ncol, FC[0][th0], FC[1][th0], FC[2][th0], FC[3][th0]);
    float bh1 = bcoef(ncol, FC[0][th1], FC[1][th1], FC[2][th1], FC[3][th1]);
    bv[c].x = lowh ? bl0 : bh0;
    bv[c].y = lowh ? bl1 : bh1;
  }
  /* per-chunk tap geometry: immediates selected by lowh (no div/mod) */
  int  adi[3][2], adj[3][2];
  bool avld[3][2];
  #pragma unroll
  for (int c = 0; c < 3; ++c){
    #pragma unroll
    for (int s = 0; s < 2; ++s){
      const int tl = 4*c + s, th = 4*c + 2 + s;
      adi[c][s]  = lowh ? TDI[tl] : TDI[th];
      adj[c][s]  = lowh ? TDJ[tl] : TDJ[th];
      avld[c][s] = lowh ? (tl < 9) : (th < 9);
    }
  }

  for (int g = 0; g < 2; ++g){
    const int p = wavebase + g*16 + rowM;
    const int b = p >> 18, i = (p >> 9) & 511, j = p & 511;
    const float* ch = blurred + ((size_t)b << 18);
    v8f acc = {0,0,0,0,0,0,0,0};
    #pragma unroll
    for (int c = 0; c < 3; ++c){
      float a0 = ldz(ch, i + adi[c][0], j + adj[c][0]);
      float a1 = ldz(ch, i + adi[c][1], j + adj[c][1]);
      v2f av;
      av.x = avld[c][0] ? a0 : 0.0f;
      av.y = avld[c][1] ? a1 : 0.0f;
      acc = __builtin_amdgcn_wmma_f32_16x16x4_f32(
              /*neg_a=*/false, av, /*neg_b=*/false, bv[c],
              /*c_mod=*/(short)0, acc, /*reuse_a=*/false, /*reuse_b=*/false);
    }
    /* D layout: VGPR r, lane<16 -> (M=r, N=lane); lane>=16 -> (M=r+8, N=lane-16) */
    #pragma unroll
    for (int r = 0; r < 8; ++r){
      int M = r + (lowh ? 0 : 8);
      sm[wave][g*16 + M][ncol] = acc[r];
    }
  }
  __syncthreads();

  const int p = wavebase + lane;
  float ex   = sm[wave][lane][0];
  float ey   = sm[wave][lane][1];
  float e45  = sm[wave][lane][2];
  float e135 = sm[wave][lane][3];
  mag[p] = sqrtf(ex*ex + ey*ey + e45*e45 + e135*e135);
  float a = atan2f(ey, ex);
  a = a - PI_F * floorf(a * (1.0f/PI_F));   /* jnp.mod(a, pi) */
  ang[p] = a;
  sob[p] = 0.5f*(fabsf(ex) + fabsf(ey));
}

/* ---------------- median via 3-pass histogram refinement ---------------- */
__global__ void k_histzero(unsigned* auxu){
  int idx = blockIdx.x*256 + threadIdx.x;
  if (idx < 8*NBINS) auxu[AX_HIST + idx] = 0u;
}

__global__ void k_hist(const float* __restrict__ sob, float* auxf){
  unsigned* auxu = (unsigned*)auxf;
  __shared__ unsigned sh[NBINS];
  for (int t = threadIdx.x; t < NBINS; t += 256) sh[t] = 0u;
  __syncthreads();
  int p = blockIdx.x*256 + threadIdx.x;
  int b = (blockIdx.x*256) >> 18;   /* block-uniform channel */
  float v  = sob[p];
  float lo = auxf[AX_LO + b], wd = auxf[AX_WD + b];
  if (v >= lo && (v - lo) < wd){
    int bin = (int)((v - lo) * ((float)NBINS / wd));
    if (bin > NBINS-1) bin = NBINS-1;
    if (bin < 0) bin = 0;
    atomicAdd(&sh[bin], 1u);
  }
  __syncthreads();
  for (int t = threadIdx.x; t < NBINS; t += 256){
    unsigned c = sh[t];
    if (c) atomicAdd(&auxu[AX_HIST + b*NBINS + t], c);
  }
}

__global__ void k_select(float* auxf){
  unsigned* auxu = (unsigned*)auxf;
  int b = threadIdx.x;
  if (b >= 8) return;
  unsigned k = auxu[AX_K + b], cum = 0u;
  int t = 0;
  for (; t < NBINS; ++t){
    unsigned c = auxu[AX_HIST + b*NBINS + t];
    if (cum + c > k) break;
    cum += c;
  }
  if (t == NBINS) t = NBINS - 1;
  auxu[AX_K + b] = k - cum;
  float wd = auxf[AX_WD + b];
  auxf[AX_LO + b] += (float)t * (wd * (1.0f/NBINS));
  auxf[AX_WD + b]  = wd * (1.0f/NBINS);
}

__global__ void k_medfinal(float* auxf){
  if (threadIdx.x != 0) return;
  float m = 0.0f;
  for (int b = 0; b < 8; ++b) m += auxf[AX_LO + b] + 0.5f*auxf[AX_WD + b];
  m *= (1.0f/8.0f);
  float s = auxf[AX_S];
  auxf[AX_MED]  = m;
  auxf[AX_LOW]  = fminf(fmaxf((1.0f - s)*m, 0.001f), 1.0f);
  auxf[AX_HIGH] = fminf(fmaxf((1.0f + s)*m, 0.001f), 1.0f);
}

/* ---------------- soft NMS + global max of sup ---------------- */
__global__ void k_sup(const float* __restrict__ mag, const float* __restrict__ ang,
                      float* __restrict__ sup, float* auxf){
  __shared__ float red[256];
  unsigned* auxu = (unsigned*)auxf;
  int p = blockIdx.x*256 + threadIdx.x;
  int b = p >> 18, i = (p >> 9) & 511, j = p & 511;
  const float* mc = mag + ((size_t)b << 18);
  float m = mc[i*512 + j];
  float a = ang[p];

  const float dirs[4] = {0.0f, PI_F*0.25f, PI_F*0.5f, PI_F*0.75f};
  float lg[4], lmax = -1e30f;
  #pragma unroll
  for (int q = 0; q < 4; ++q){
    float d = fabsf(a - dirs[q]);
    d = fminf(d, PI_F - d);
    lg[q] = -d * 100.0f;             /* /0.01 */
    lmax = fmaxf(lmax, lg[q]);
  }
  float lsum = 0.0f;
  #pragma unroll
  for (int q = 0; q < 4; ++q){ lg[q] = expf(lg[q] - lmax); lsum += lg[q]; }
  float inv = 1.0f / lsum;

  int im = clampi(i-1,0,511), ip = clampi(i+1,0,511);
  int jm = clampi(j-1,0,511), jp = clampi(j+1,0,511);
  float n1[4], n2[4];
  n1[0] = mc[i*512 + jm];  n2[0] = mc[i*512 + jp];
  n1[1] = mc[im*512 + jm]; n2[1] = mc[ip*512 + jp];
  n1[2] = mc[im*512 + j];  n2[2] = mc[ip*512 + j];
  n1[3] = mc[im*512 + jp]; n2[3] = mc[ip*512 + jm];

  float sv = 0.0f;
  #pragma unroll
  for (int q = 0; q < 4; ++q){
    float mx3 = fmaxf(m, fmaxf(n1[q], n2[q]));
    float e0 = expf((m      - mx3)*10.0f);
    float e1 = expf((n1[q]  - mx3)*10.0f);
    float e2 = expf((n2[q]  - mx3)*10.0f);
    sv += m * (e0/(e0+e1+e2)) * lg[q] * inv;
  }
  sup[p] = sv;

  red[threadIdx.x] = sv; __syncthreads();
  for (int st = 128; st > 0; st >>= 1){
    if (threadIdx.x < st) red[threadIdx.x] = fmaxf(red[threadIdx.x], red[threadIdx.x+st]);
    __syncthreads();
  }
  if (threadIdx.x == 0) atomicMax(&auxu[AX_MXKEY], okey(red[0]));
}

/* ---------------- hysteresis result + sobel enhance + min/max ---------------- */
__global__ void k_result(const float* __restrict__ sup, const float* __restrict__ img,
                         float* __restrict__ res, float* auxf){
  __shared__ float rmn[256], rmx[256];
  unsigned* auxu = (unsigned*)auxf;
  int p = blockIdx.x*256 + threadIdx.x;
  int b = p >> 18, i = (p >> 9) & 511, j = p & 511;
  float mx   = keyf(auxu[AX_MXKEY]);
  float idn  = 1.0f / ((mx > 0.0f) ? mx : 1.0f);
  float lo   = auxf[AX_LOW], hi = auxf[AX_HIGH], wgt = auxf[AX_W];
  const float* sc = sup + ((size_t)b << 18);
  const float* ic = img + ((size_t)b << 18);

  float sc_c = sc[i*512 + j] * idn;
  float Sh = hsw(sc_c - hi), Sl = hsw(sc_c - lo);
  float dil = 0.0f;                       /* conv(Sh, ones3x3), zero padded */
  #pragma unroll
  for (int a = -1; a <= 1; ++a){
    #pragma unroll
    for (int c = -1; c <= 1; ++c){
      int ii = i + a, jj = j + c;
      float sv = sc[clampi(ii,0,511)*512 + clampi(jj,0,511)] * idn;
      bool inr = ((unsigned)ii < 512u) && ((unsigned)jj < 512u);
      dil += inr ? hsw(sv - hi) : 0.0f;
    }
  }
  float r = sc_c*(Sh + Sl) + dil*Sl*(1.0f - Sh);

  /* sobel of ORIGINAL input, zero padding, branchless; taps from constexpr */
  float ex = 0.0f, ey = 0.0f;
  #pragma unroll
  for (int t = 0; t < 9; ++t){
    float xv = ldz(ic, i + TDI[t], j + TDJ[t]);
    ex += FC[0][t]*xv; ey += FC[1][t]*xv;
  }
  float sb = 0.5f*(fabsf(ex) + fabsf(ey));
  r = (1.0f - wgt)*r + wgt*sb;
  res[p] = r;

  rmn[threadIdx.x] = r; rmx[threadIdx.x] = r; __syncthreads();
  for (int st = 128; st > 0; st >>= 1){
    if (threadIdx.x < st){
      rmn[threadIdx.x] = fminf(rmn[threadIdx.x], rmn[threadIdx.x+st]);
      rmx[threadIdx.x] = fmaxf(rmx[threadIdx.x], rmx[threadIdx.x+st]);
    }
    __syncthreads();
  }
  if (threadIdx.x == 0){
    atomicMin(&auxu[AX_MINKEY], okey(rmn[0]));
    atomicMax(&auxu[AX_MAXKEY], okey(rmx[0]));
  }
}

__global__ void k_norm(const float* __restrict__ res, float* __restrict__ out,
                       const float* auxf){
  const unsigned* auxu = (const unsigned*)auxf;
  int p = blockIdx.x*256 + threadIdx.x;
  float rmin = keyf(auxu[AX_MINKEY]), rmax = keyf(auxu[AX_MAXKEY]);
  out[p] = (res[p] - rmin) / (rmax - rmin);
}

__global__ void k_combine(float* __restrict__ out){
  int p = blockIdx.x*256 + threadIdx.x;
  float m1 = out[(size_t)NTOT + p];
  float m2 = out[(size_t)2*NTOT + p];
  float f = (1.0f - m1)*(1.0f - m2) + m1*m2;
  out[p] = 1.0f - f;
}

extern "C" void kernel_launch(void* const* d_in, const int* in_sizes, int n_in,
                              void* d_out, int out_size, void* d_ws, size_t ws_size,
                              hipStream_t stream){
  (void)in_sizes; (void)n_in; (void)out_size; (void)ws_size;
  const float* imgs[2] = {(const float*)d_in[0], (const float*)d_in[1]};
  const float* sig[2]  = {(const float*)d_in[2], (const float*)d_in[4]};
  const float* wgt[2]  = {(const float*)d_in[3], (const float*)d_in[5]};
  float* out  = (float*)d_out;
  char*  ws   = (char*)d_ws;
  float* auxf = (float*)ws;
  float* P0 = (float*)(ws + BIGOFF);
  float* P1 = P0 + NTOT;
  float* P2 = P1 + NTOT;
  float* P3 = P2 + NTOT;

  dim3 grid(NTOT/256), blk(256);

  for (int im = 0; im < 2; ++im){
    const float* img = imgs[im];
    k_prep<<<1, 64, 0, stream>>>(sig[im], wgt[im], auxf);

    /* adaptive smoothing: 3 iterations, ping-pong */
    k_smooth_w<<<grid, blk, 0, stream>>>(img, P2);
    k_smooth_f<<<grid, blk, 0, stream>>>(img, P2, P0);
    k_smooth_w<<<grid, blk, 0, stream>>>(P0,  P2);
    k_smooth_f<<<grid, blk, 0, stream>>>(P0,  P2, P1);
    k_smooth_w<<<grid, blk, 0, stream>>>(P1,  P2);
    k_smooth_f<<<grid, blk, 0, stream>>>(P1,  P2, P0);   /* blurred -> P0 */

    /* WMMA directional conv bank: mag->P1, ang->P2, sob->P3 */
    k_dir_wmma<<<grid, blk, 0, stream>>>(P0, P1, P2, P3);

    /* per-channel median of sob via 3-pass histogram refinement */
    for (int pass = 0; pass < 3; ++pass){
      k_histzero<<<64, 256, 0, stream>>>((unsigned*)auxf);
      k_hist<<<grid, blk, 0, stream>>>(P3, auxf);
      k_select<<<1, 8, 0, stream>>>(auxf);
    }
    k_medfinal<<<1, 1, 0, stream>>>(auxf);

    /* soft NMS (+ global max): sup -> P0 */
    k_sup<<<grid, blk, 0, stream>>>(P1, P2, P0, auxf);

    /* hysteresis + enhance (+ global min/max): res -> P1 */
    k_result<<<grid, blk, 0, stream>>>(P0, img, P1, auxf);

    /* normalize -> m_i straight into its output slot */
    k_norm<<<grid, blk, 0, stream>>>(P1, out + (size_t)(1 + im)*NTOT, auxf);
  }

  k_combine<<<grid, blk, 0, stream>>>(out);
}